// KNNMask_32169305047733
// MI455X (gfx1250) — compile-verified
//
#include <hip/hip_runtime.h>

// ---------------------------------------------------------------------------
// KNN top-K mask, K=32, rows of 16384 fp32.
// One workgroup per row. Row staged to LDS once (CDNA5 async global->LDS,
// GLOBAL_LOAD_ASYNC_TO_LDS_B128 + s_wait_asynccnt), exact K-th-largest found
// by 4-pass MSB-first radix select on monotone keys, mask written back in one
// coalesced pass. HBM traffic = 1 read + 1 write (roofline ~44us @ 23.3TB/s).
// ---------------------------------------------------------------------------

#define K_SEL   32
#define ROW_LEN 16384
#define BLOCK   256
#define NWAVES  (BLOCK / 32)                    // 8 waves (wave32)
#define CHUNKS  (ROW_LEN / 4 / BLOCK)           // 16 float4 chunks per thread

#if defined(__AMDGCN__) && __has_builtin(__builtin_amdgcn_global_load_async_to_lds_b128)
#define USE_ASYNC_LDS 1
#else
#define USE_ASYNC_LDS 0
#endif

// The builtin's parameters are pointers to a 4 x i32 vector (per hipcc
// diagnostic), generic address space.
typedef int v4i __attribute__((ext_vector_type(4)));

// Monotone map: key(a) < key(b)  <=>  a < b  (as floats, no NaNs in input)
__device__ __forceinline__ unsigned key_of(unsigned bits) {
    return (bits & 0x80000000u) ? ~bits : (bits | 0x80000000u);
}

__global__ void __launch_bounds__(BLOCK)
knn_mask_kernel(const float* __restrict__ sim, float* __restrict__ out) {
    const int row = blockIdx.x;
    const int tid = threadIdx.x;

    __shared__ __align__(16) unsigned rowbits[ROW_LEN];      // 64 KB raw float bits
    __shared__ unsigned hist[NWAVES][256];                   // 8 KB privatized bins
    __shared__ int s_chosen, s_want, s_ticket;

    const float* rowPtr = sim + (size_t)row * ROW_LEN;

    // ---- Stage row into LDS ------------------------------------------------
#if USE_ASYNC_LDS
    for (int j = 0; j < CHUNKS; ++j) {
        const int c = j * BLOCK + tid;                       // float4 chunk id
        __builtin_amdgcn_global_load_async_to_lds_b128(
            (v4i*)(rowPtr + 4 * c),
            (v4i*)(&rowbits[4 * c]),
            /*offset=*/0, /*cpol=*/0);
    }
#if __has_builtin(__builtin_amdgcn_s_wait_asynccnt)
    __builtin_amdgcn_s_wait_asynccnt(0);
#else
    asm volatile("s_wait_asynccnt 0" ::: "memory");
#endif
#else
    {
        const uint4* g = (const uint4*)rowPtr;
        uint4* s = (uint4*)rowbits;
        for (int j = 0; j < CHUNKS; ++j) {
            const int c = j * BLOCK + tid;
            s[c] = g[c];
        }
    }
#endif
    if (tid == 0) s_ticket = 0;
    __syncthreads();

    // ---- 4-pass MSB-first radix select for the K-th largest key -----------
    unsigned prefix = 0u, mask = 0u;
    int want = K_SEL;                      // how many still to take at/below prefix
    const uint4* srow = (const uint4*)rowbits;
    unsigned* myhist = hist[tid >> 5];

    for (int shift = 24; shift >= 0; shift -= 8) {
        // clear privatized histograms
        for (int b = tid; b < NWAVES * 256; b += BLOCK)
            ((unsigned*)hist)[b] = 0u;
        __syncthreads();

        for (int j = 0; j < CHUNKS; ++j) {
            const uint4 v = srow[j * BLOCK + tid];
            const unsigned k0 = key_of(v.x), k1 = key_of(v.y);
            const unsigned k2 = key_of(v.z), k3 = key_of(v.w);
            if ((k0 & mask) == prefix) atomicAdd(&myhist[(k0 >> shift) & 255u], 1u);
            if ((k1 & mask) == prefix) atomicAdd(&myhist[(k1 >> shift) & 255u], 1u);
            if ((k2 & mask) == prefix) atomicAdd(&myhist[(k2 >> shift) & 255u], 1u);
            if ((k3 & mask) == prefix) atomicAdd(&myhist[(k3 >> shift) & 255u], 1u);
        }
        __syncthreads();

        if (tid == 0) {
            int w = want, chosen = 0;
            for (int b = 255; b >= 0; --b) {
                int cnt = 0;
                for (int wv = 0; wv < NWAVES; ++wv) cnt += (int)hist[wv][b];
                if (cnt >= w) { chosen = b; break; }
                w -= cnt;
            }
            s_chosen = chosen;
            s_want = w;
        }
        __syncthreads();

        prefix |= ((unsigned)s_chosen) << shift;
        mask   |= 0xFFu << shift;
        want = s_want;
    }

    const unsigned T = prefix;             // exact key of the K-th largest element
    const int need = want;                 // how many key==T elements to select

    // ---- Emit mask: 0 at top-K, +inf elsewhere ----------------------------
    const float INF = __builtin_inff();
    float4* orow = (float4*)(out + (size_t)row * ROW_LEN);
    for (int j = 0; j < CHUNKS; ++j) {
        const int c = j * BLOCK + tid;
        const uint4 v = srow[c];
        float4 o;
        {
            const unsigned k = key_of(v.x);
            o.x = (k > T) ? 0.0f
                : (k == T && atomicAdd(&s_ticket, 1) < need) ? 0.0f : INF;
        }
        {
            const unsigned k = key_of(v.y);
            o.y = (k > T) ? 0.0f
                : (k == T && atomicAdd(&s_ticket, 1) < need) ? 0.0f : INF;
        }
        {
            const unsigned k = key_of(v.z);
            o.z = (k > T) ? 0.0f
                : (k == T && atomicAdd(&s_ticket, 1) < need) ? 0.0f : INF;
        }
        {
            const unsigned k = key_of(v.w);
            o.w = (k > T) ? 0.0f
                : (k == T && atomicAdd(&s_ticket, 1) < need) ? 0.0f : INF;
        }
        orow[c] = o;
    }
}

extern "C" void kernel_launch(void* const* d_in, const int* in_sizes, int n_in,
                              void* d_out, int out_size, void* d_ws, size_t ws_size,
                              hipStream_t stream) {
    (void)n_in; (void)d_ws; (void)ws_size; (void)out_size;
    const float* sim = (const float*)d_in[0];
    float* out = (float*)d_out;
    const int rows = in_sizes[0] / ROW_LEN;   // 8192
    knn_mask_kernel<<<dim3(rows), dim3(BLOCK), 0, stream>>>(sim, out);
}